// metaBNSTModelSTAGIN_57105885168079
// MI455X (gfx1250) — compile-verified
//
#include <hip/hip_runtime.h>
#include <hip/hip_bf16.h>

typedef float v2f __attribute__((ext_vector_type(2)));
typedef float v8f __attribute__((ext_vector_type(8)));

#define N_NODES 50000
#define N_EDGES 800000
#define D_IN    128
#define D_HID   256
#define D_OUTD  128
#define BN_EPS  1e-5f

__device__ __forceinline__ void atomAddF32(float* p, float v) {
    // lowers to global_atomic_add_f32 (no CAS loop)
    unsafeAtomicAdd(p, v);
}

// ---------------------------------------------------------------------------
// 0) zero a small float buffer (BN stats accumulators)
// ---------------------------------------------------------------------------
__global__ void zero_kernel(float* __restrict__ p, int n) {
    int i = blockIdx.x * blockDim.x + threadIdx.x;
    if (i < n) p[i] = 0.0f;
}

// ---------------------------------------------------------------------------
// 1) agg = epsilon * v   (epsilon is a 1-element device array)
// ---------------------------------------------------------------------------
__global__ void init_agg_kernel(const float* __restrict__ v,
                                const float* __restrict__ epsilon,
                                float* __restrict__ agg, int n) {
    int i = blockIdx.x * blockDim.x + threadIdx.x;
    float eps = epsilon[0];
    if (i < n) agg[i] = eps * v[i];
}

// ---------------------------------------------------------------------------
// 2) edge scatter: agg[r,:] += w_e * v[c,:]
//    one wave (32 lanes) per edge; lane handles a float4 slice of the 128-dim
//    row -> coalesced B128 gather of v + 4 global fp32 atomics per lane.
// ---------------------------------------------------------------------------
__global__ void scatter_kernel(const float* __restrict__ v,
                               const int*   __restrict__ erows,
                               const int*   __restrict__ ecols,
                               const float* __restrict__ evals,
                               float* __restrict__ agg) {
    long long t = (long long)blockIdx.x * blockDim.x + threadIdx.x;
    int e = (int)(t >> 5);
    if (e >= N_EDGES) return;
    int f = ((int)t & 31) << 2;                 // feature offset 0..124
    int r = erows[e];
    int c = ecols[e];
    float w = evals[e];
    const float4 vv = *(const float4*)(v + (long long)c * D_IN + f);
    float* dst = agg + (long long)r * D_IN + f;
    atomAddF32(dst + 0, w * vv.x);
    atomAddF32(dst + 1, w * vv.y);
    atomAddF32(dst + 2, w * vv.z);
    atomAddF32(dst + 3, w * vv.w);
}

// ---------------------------------------------------------------------------
// 3) C[M,Nc] = A[M,K] @ W[K,Nc] + bias, via V_WMMA_F32_16X16X4_F32.
//    Block = 256 threads = 8 waves. blockIdx.x -> 16-row stripe,
//    blockIdx.y*8 + wave -> 16-col tile. A tile staged in LDS (pad=4 floats
//    per row => bank-conflict-free ds_load_b64 fragments).
//
//    ISA VGPR layouts (CDNA5 ISA §7.12.2):
//      A 16x4 f32 : lane L -> row L%16, K = k + 2*(L>=16) + {0,1}   (2 VGPRs)
//      B 4x16 f32 : lane L -> col L%16, K = k + 2*(L>=16) + {0,1}   (2 VGPRs)
//      C/D 16x16  : VGPR i, lane L -> row i + 8*(L>=16), col L%16   (8 VGPRs)
// ---------------------------------------------------------------------------
__global__ void gemm_bias_kernel(const float* __restrict__ A,
                                 const float* __restrict__ W,
                                 const float* __restrict__ bias,
                                 float* __restrict__ C,
                                 int M, int K, int Nc) {
    __shared__ float As[16 * (256 + 4)];        // max K = 256, pad 4
    const int KP      = K + 4;
    const int rowbase = blockIdx.x << 4;
    const int wave    = threadIdx.x >> 5;
    const int lane    = threadIdx.x & 31;
    const int colbase = ((blockIdx.y << 3) + wave) << 4;

    // cooperative load of the 16 x K A-tile (float4 granularity, coalesced)
    const int vecPerRow = K >> 2;
    for (int idx = threadIdx.x; idx < 16 * vecPerRow; idx += blockDim.x) {
        int r  = idx / vecPerRow;
        int kc = (idx - r * vecPerRow) << 2;
        float4 t = *(const float4*)(A + (long long)(rowbase + r) * K + kc);
        float* s = &As[r * KP + kc];
        s[0] = t.x; s[1] = t.y; s[2] = t.z; s[3] = t.w;
    }
    __syncthreads();

    v8f acc = {};
    const int m     = lane & 15;
    const int khalf = (lane >> 4) << 1;         // 0 or 2
    const int ncol  = colbase + (lane & 15);

    #pragma unroll 8
    for (int k = 0; k < K; k += 4) {
        int k0 = k + khalf;
        v2f a = *(const v2f*)&As[m * KP + k0];  // 8B-aligned ds_load_b64
        v2f b;
        b.x = W[(long long)k0       * Nc + ncol];
        b.y = W[(long long)(k0 + 1) * Nc + ncol];
        acc = __builtin_amdgcn_wmma_f32_16x16x4_f32(
                  false, a, false, b, (short)0, acc, false, false);
    }

    const int rbase = rowbase + ((lane >> 4) << 3);   // +8 for lanes 16..31
    const float bv  = bias[ncol];
    #pragma unroll
    for (int i = 0; i < 8; ++i) {
        C[(long long)(rbase + i) * Nc + ncol] = acc[i] + bv;
    }
}

// ---------------------------------------------------------------------------
// 4) per-column partial sums / sum-of-squares (block = C threads; each block
//    owns a row slab; fully coalesced reads; 2 fp32 atomics per column/block)
// ---------------------------------------------------------------------------
__global__ void colstats_kernel(const float* __restrict__ x,
                                float* __restrict__ sums,
                                float* __restrict__ sqs,
                                int Nrows, int C, int rowsPerBlock) {
    int c  = threadIdx.x;
    int r0 = blockIdx.x * rowsPerBlock;
    int r1 = min(r0 + rowsPerBlock, Nrows);
    float s = 0.0f, s2 = 0.0f;
    for (int r = r0; r < r1; ++r) {
        float v = x[(long long)r * C + c];
        s  += v;
        s2 += v * v;
    }
    atomAddF32(&sums[c], s);
    atomAddF32(&sqs[c],  s2);
}

// ---------------------------------------------------------------------------
// 5) in-place BN(train, biased var) + affine + ReLU
// ---------------------------------------------------------------------------
__global__ void bnrelu_kernel(float* __restrict__ x,
                              const float* __restrict__ sums,
                              const float* __restrict__ sqs,
                              const float* __restrict__ gamma,
                              const float* __restrict__ beta,
                              int total, int Cmask, float invN) {
    int i = blockIdx.x * blockDim.x + threadIdx.x;
    if (i >= total) return;
    int c = i & Cmask;                          // C is a power of two
    float mean = sums[c] * invN;
    float var  = sqs[c] * invN - mean * mean;
    float xh   = (x[i] - mean) * rsqrtf(var + BN_EPS);
    float y    = xh * gamma[c] + beta[c];
    x[i] = fmaxf(y, 0.0f);
}

// ---------------------------------------------------------------------------
extern "C" void kernel_launch(void* const* d_in, const int* in_sizes, int n_in,
                              void* d_out, int out_size, void* d_ws, size_t ws_size,
                              hipStream_t stream) {
    const float* v     = (const float*)d_in[0];
    const int*   erows = (const int*)  d_in[1];
    const int*   ecols = (const int*)  d_in[2];
    const float* evals = (const float*)d_in[3];
    const float* w1    = (const float*)d_in[4];
    const float* b1    = (const float*)d_in[5];
    const float* g1    = (const float*)d_in[6];
    const float* be1   = (const float*)d_in[7];
    const float* w2    = (const float*)d_in[8];
    const float* b2    = (const float*)d_in[9];
    const float* g2    = (const float*)d_in[10];
    const float* be2   = (const float*)d_in[11];
    const float* eps   = (const float*)d_in[12];

    char*  ws    = (char*)d_ws;
    float* agg   = (float*)ws;                                         // N*128 f32
    float* z1    = (float*)(ws + (size_t)N_NODES * D_IN * 4);          // N*256 f32
    float* stats = (float*)(ws + (size_t)N_NODES * D_IN * 4
                               + (size_t)N_NODES * D_HID * 4);         // 768 f32
    float* sums1 = stats;
    float* sqs1  = stats + 256;
    float* sums2 = stats + 512;
    float* sqs2  = stats + 640;
    float* outp  = (float*)d_out;

    // zero BN stat accumulators (workspace is poisoned, and graph replays
    // must not accumulate across calls)
    zero_kernel<<<3, 256, 0, stream>>>(stats, 768);

    // agg = eps * v
    {
        int n = N_NODES * D_IN;
        init_agg_kernel<<<(n + 255) / 256, 256, 0, stream>>>(v, eps, agg, n);
    }

    // sparse scatter: one wave per edge
    {
        long long threads = (long long)N_EDGES * 32;
        scatter_kernel<<<(unsigned)((threads + 255) / 256), 256, 0, stream>>>(
            v, erows, ecols, evals, agg);
    }

    // layer 1: z1 = agg @ w1 + b1 ; BN stats ; BN+ReLU in place
    gemm_bias_kernel<<<dim3(N_NODES / 16, D_HID / 128), 256, 0, stream>>>(
        agg, w1, b1, z1, N_NODES, D_IN, D_HID);
    colstats_kernel<<<(N_NODES + 127) / 128, D_HID, 0, stream>>>(
        z1, sums1, sqs1, N_NODES, D_HID, 128);
    {
        int total = N_NODES * D_HID;
        bnrelu_kernel<<<(total + 255) / 256, 256, 0, stream>>>(
            z1, sums1, sqs1, g1, be1, total, D_HID - 1, 1.0f / N_NODES);
    }

    // layer 2: out = z1 @ w2 + b2 ; BN stats ; BN+ReLU in place (on d_out)
    gemm_bias_kernel<<<dim3(N_NODES / 16, D_OUTD / 128), 256, 0, stream>>>(
        z1, w2, b2, outp, N_NODES, D_HID, D_OUTD);
    colstats_kernel<<<(N_NODES + 127) / 128, D_OUTD, 0, stream>>>(
        outp, sums2, sqs2, N_NODES, D_OUTD, 128);
    {
        int total = N_NODES * D_OUTD;
        bnrelu_kernel<<<(total + 255) / 256, 256, 0, stream>>>(
            outp, sums2, sqs2, g2, be2, total, D_OUTD - 1, 1.0f / N_NODES);
    }
}